// LIFSpike_73864847556979
// MI455X (gfx1250) — compile-verified
//
#include <hip/hip_runtime.h>

// LIF spike recurrence, memory-bound streaming kernel for MI455X (gfx1250).
// x: [B,C,H,W,T] f32 with T=8 innermost; out: spikes, same shape/dtype.
// 268 MB total traffic @ 23.3 TB/s -> ~11.5 us floor; compute is negligible.

#define LIF_TAU 0.2f
#define LIF_VTH 0.3f
#define T_STEPS 8
#define BLOCK   256   // 8 wave32s per block

// Native clang vector type: valid for __builtin_nontemporal_* and lowers to
// b128 global/ds memory ops.
typedef float v4f __attribute__((ext_vector_type(4)));

__global__ __launch_bounds__(BLOCK)
void lif_spike_kernel(const float* __restrict__ x,
                      float* __restrict__ out,
                      int n_pixels) {
    // 32-byte staging slot per lane: 8 KB LDS per 256-thread block.
    __shared__ v4f tile[2 * BLOCK];

    const int tid = threadIdx.x;
    const int idx = blockIdx.x * BLOCK + tid;
    if (idx >= n_pixels) return;

    const v4f* gsrc = reinterpret_cast<const v4f*>(x) + 2 * (size_t)idx;

    // LDS byte offset of this lane's staging slot (addrspace(3) offset is the
    // low 32 bits of the generic pointer).
    unsigned lds_off = (unsigned)(uintptr_t)(&tile[2 * tid]);

    // CDNA5 async global->LDS copies (ASYNCcnt-tracked), 2 x b128 per lane
    // = this lane's 8 contiguous timesteps. offset:16 applies to both the
    // global and LDS addresses per the ISA address formulas.
    asm volatile(
        "global_load_async_to_lds_b128 %0, %1, off\n\t"
        "global_load_async_to_lds_b128 %0, %1, off offset:16\n\t"
        "s_wait_asynccnt 0x0"
        :
        : "v"(lds_off), "v"(gsrc)
        : "memory");

    // Read back from LDS (ds_load_b128 x2). Same-lane producer/consumer, so
    // the per-wave asynccnt wait is sufficient ordering; no barrier needed.
    v4f a = tile[2 * tid];
    v4f b = tile[2 * tid + 1];
    float xv[T_STEPS] = {a.x, a.y, a.z, a.w, b.x, b.y, b.z, b.w};

    // Serial LIF recurrence over T (fully unrolled, branchless select).
    float u = 0.0f, o = 0.0f;
    float ov[T_STEPS];
#pragma unroll
    for (int t = 0; t < T_STEPS; ++t) {
        u = LIF_TAU * u * (1.0f - o) + xv[t];
        o = ((u - LIF_VTH) > 0.0f) ? 1.0f : 0.0f;   // match reference exactly
        ov[t] = o;
    }

    // Non-temporal 128-bit stores: single-pass stream, don't dirty caches.
    v4f r0 = {ov[0], ov[1], ov[2], ov[3]};
    v4f r1 = {ov[4], ov[5], ov[6], ov[7]};
    v4f* gdst = reinterpret_cast<v4f*>(out) + 2 * (size_t)idx;
    __builtin_nontemporal_store(r0, gdst);
    __builtin_nontemporal_store(r1, gdst + 1);
}

extern "C" void kernel_launch(void* const* d_in, const int* in_sizes, int n_in,
                              void* d_out, int out_size, void* d_ws, size_t ws_size,
                              hipStream_t stream) {
    (void)n_in; (void)out_size; (void)d_ws; (void)ws_size;

    const float* x   = (const float*)d_in[0];
    float*       out = (float*)d_out;

    // Elements = B*C*H*W*T; each lane owns one pixel's T=8 chain.
    const int n_pixels = in_sizes[0] / T_STEPS;   // 4,194,304
    const int blocks   = (n_pixels + BLOCK - 1) / BLOCK;

    lif_spike_kernel<<<blocks, BLOCK, 0, stream>>>(x, out, n_pixels);
}